// Conv_Decomp_7258494730848
// MI455X (gfx1250) — compile-verified
//
#include <hip/hip_runtime.h>

typedef __attribute__((ext_vector_type(2))) float v2f;
typedef __attribute__((ext_vector_type(8))) float v8f;

#define NUM_FILTERS 128
#define FH 3
#define FW 3
#define C_IN 64
#define RANK 64
#define H_IN 256
#define W_IN 256
#define HO (H_IN - FH + 1)          // 254
#define WO (W_IN - FW + 1)          // 254
#define NPIX (H_IN * W_IN)          // 65536
#define NOUT (HO * WO)              // 64516

// ---------------------------------------------------------------------------
// Kernel 1: rank projection  T1[r, p] = sum_c factor3[c, r] * input[c, p]
//   GEMM M=64 (rank), K=64 (channels), N=65536 (pixels)
//   One wave -> 16 pixels x 64 ranks (4 WMMA M-strips, 16 k-steps of 4).
// ---------------------------------------------------------------------------
__global__ void __launch_bounds__(256)
cp_rank_project(const float* __restrict__ input,
                const float* __restrict__ f3,      // (C, RANK) row-major
                float* __restrict__ t1)            // (RANK, NPIX)
{
    const int lane  = threadIdx.x & 31;            // wave32
    const int wave  = threadIdx.x >> 5;
    const int pix0  = (blockIdx.x * 8 + wave) * 16;
    const int col   = lane & 15;                   // N index within tile
    const int khalf = lane >> 4;                   // 0 or 1

    v8f acc[4] = {};

    for (int kb = 0; kb < C_IN; kb += 4) {
        const int c0 = kb + 2 * khalf;
        // B fragment: B[k][n] = input[c0+k'][pix0+col]  (coalesced b32 loads)
        v2f b;
        b.x = input[(long)c0       * NPIX + pix0 + col];
        b.y = input[(long)(c0 + 1) * NPIX + pix0 + col];
#pragma unroll
        for (int s = 0; s < 4; ++s) {
            // A fragment: A[m][k] = factor3[c][r],  m = rank, k = channel
            const int r = s * 16 + col;
            v2f a;
            a.x = f3[(long)c0       * RANK + r];
            a.y = f3[(long)(c0 + 1) * RANK + r];
            acc[s] = __builtin_amdgcn_wmma_f32_16x16x4_f32(
                false, a, false, b, (short)0, acc[s], false, false);
        }
    }

#pragma unroll
    for (int s = 0; s < 4; ++s)
#pragma unroll
        for (int i = 0; i < 8; ++i) {
            const int r = s * 16 + i + 8 * khalf;  // C/D layout: vgpr i -> row i / i+8
            t1[(long)r * NPIX + pix0 + col] = acc[s][i];
        }
}

// ---------------------------------------------------------------------------
// Kernel 2: separable 3x3 tap per rank (tiny FLOP count -> plain VALU)
//   T3[r, y, x] = sum_{h,w} T1[r, y+h, x+w] * f1[h,r] * f2[w,r]
//   grid: x covers output pixels, y covers rank (no integer division).
// ---------------------------------------------------------------------------
__global__ void __launch_bounds__(256)
cp_sep_conv(const float* __restrict__ t1,
            const float* __restrict__ f1,          // (FH, RANK)
            const float* __restrict__ f2,          // (FW, RANK)
            float* __restrict__ t3)                // (RANK, NOUT)
{
    const int p = blockIdx.x * blockDim.x + threadIdx.x;
    const int r = blockIdx.y;
    if (p >= NOUT) return;
    const int y = p / WO;
    const int x = p - y * WO;

    const float w10 = f1[0 * RANK + r], w11 = f1[1 * RANK + r], w12 = f1[2 * RANK + r];
    const float w20 = f2[0 * RANK + r], w21 = f2[1 * RANK + r], w22 = f2[2 * RANK + r];

    const float* base = t1 + (long)r * NPIX + y * W_IN + x;
    const float row0 = base[0]          * w20 + base[1]            * w21 + base[2]            * w22;
    const float row1 = base[W_IN]       * w20 + base[W_IN + 1]     * w21 + base[W_IN + 2]     * w22;
    const float row2 = base[2 * W_IN]   * w20 + base[2 * W_IN + 1] * w21 + base[2 * W_IN + 2] * w22;

    t3[(long)r * NOUT + p] = row0 * w10 + row1 * w11 + row2 * w12;
}

// ---------------------------------------------------------------------------
// Kernel 3: filter expansion  out[f, n] = sum_r factor0[f, r] * T3[r, n]
//   GEMM M=128 (filters), K=64 (rank), N=64516 (pixels)
//   One wave -> 16 pixels x 128 filters (8 WMMA M-strips).
//   Edge tile: clamp B-load addresses (GEMM columns are independent; the
//   clamped columns are never stored), keeping EXEC all-ones for WMMA.
// ---------------------------------------------------------------------------
__global__ void __launch_bounds__(256)
cp_filter_expand(const float* __restrict__ t3,
                 const float* __restrict__ f0,     // (NUM_FILTERS, RANK)
                 float* __restrict__ out)          // (NUM_FILTERS, NOUT)
{
    const int lane  = threadIdx.x & 31;
    const int wave  = threadIdx.x >> 5;
    const int tile  = blockIdx.x * 8 + wave;
    const int n0    = tile * 16;
    if (n0 >= NOUT) return;                        // whole-wave uniform exit

    const int col    = lane & 15;
    const int khalf  = lane >> 4;
    const int n      = n0 + col;
    const int nclamp = n < NOUT ? n : (NOUT - 1);

    v8f acc[8] = {};

    for (int kb = 0; kb < RANK; kb += 4) {
        const int k0 = kb + 2 * khalf;
        v2f b;
        b.x = t3[(long)k0       * NOUT + nclamp];
        b.y = t3[(long)(k0 + 1) * NOUT + nclamp];
#pragma unroll
        for (int s = 0; s < 8; ++s) {
            const int f = s * 16 + col;
            // contiguous (k0, k0+1): single 8-byte aligned load
            const v2f a = *(const v2f*)(f0 + (long)f * RANK + k0);
            acc[s] = __builtin_amdgcn_wmma_f32_16x16x4_f32(
                false, a, false, b, (short)0, acc[s], false, false);
        }
    }

    if (n < NOUT) {
#pragma unroll
        for (int s = 0; s < 8; ++s)
#pragma unroll
            for (int i = 0; i < 8; ++i) {
                const int f = s * 16 + i + 8 * khalf;
                out[(long)f * NOUT + n] = acc[s][i];
            }
    }
}

// ---------------------------------------------------------------------------
extern "C" void kernel_launch(void* const* d_in, const int* in_sizes, int n_in,
                              void* d_out, int out_size, void* d_ws, size_t ws_size,
                              hipStream_t stream) {
    const float* input = (const float*)d_in[0];    // (C, H, W)
    const float* f0    = (const float*)d_in[1];    // (NUM_FILTERS, RANK)
    const float* f1    = (const float*)d_in[2];    // (FH, RANK)
    const float* f2    = (const float*)d_in[3];    // (FW, RANK)
    const float* f3    = (const float*)d_in[4];    // (C, RANK)
    float*       out   = (float*)d_out;            // (NUM_FILTERS, HO, WO)

    float* t1 = (float*)d_ws;                      // RANK*NPIX floats (16.8 MB)
    float* t3 = t1 + (size_t)RANK * NPIX;          // RANK*NOUT floats (16.5 MB)

    // K1: 65536 pixels / (8 waves * 16 px) = 512 blocks
    cp_rank_project<<<512, 256, 0, stream>>>(input, f3, t1);

    // K2: pixels x rank grid
    dim3 g2((NOUT + 255) / 256, RANK);
    cp_sep_conv<<<g2, 256, 0, stream>>>(t1, f1, f2, t3);

    // K3: ceil(64516/16) = 4033 n-tiles, 8 waves/block -> 505 blocks
    cp_filter_expand<<<505, 256, 0, stream>>>(t3, f0, out);
}